// Model_58488864637193
// MI455X (gfx1250) — compile-verified
//
#include <hip/hip_runtime.h>
#include <hip/hip_bf16.h>

typedef __attribute__((ext_vector_type(2))) float v2f;
typedef __attribute__((ext_vector_type(8))) float v8f;

// Sizes fixed by the reference
#define L_TOK   4096
#define B_BATCH 8
#define K0      484     // 121*4
#define H0      32
#define TB      128     // tokens per block in stage1
#define HALO    120     // WOW1-1

// ---------------- Precompute 1: Wsr = Ws(512x4096) @ Wr(4096x4) ----------------
__global__ __launch_bounds__(256) void k_wsr(const float* __restrict__ Ws,
                                             const float* __restrict__ Wr,
                                             float* __restrict__ Wsr) {
    int id = blockIdx.x * 256 + threadIdx.x;   // 2048 threads: 512 rows x 4 cols
    int r = id >> 2, o = id & 3;
    const float* wsrow = Ws + (size_t)r * 4096;
    float acc = 0.f;
    for (int k = 0; k < 4096; ++k) acc = fmaf(wsrow[k], Wr[k * 4 + o], acc);
    Wsr[r * 4 + o] = acc;
}

// ------------- Precompute 2: Wc = W1(32x512)@Wsr ; bc = b1@Wsr + bs@Wr + br -------------
__global__ __launch_bounds__(256) void k_fold(const float* __restrict__ W1,
                                              const float* __restrict__ b1,
                                              const float* __restrict__ bs,
                                              const float* __restrict__ Wr,
                                              const float* __restrict__ br,
                                              const float* __restrict__ Wsr,
                                              float* __restrict__ Wc,
                                              float* __restrict__ bc) {
    int tid = threadIdx.x;
    if (tid < 128) {                       // Wc[n][o]
        int n = tid >> 2, o = tid & 3;
        float acc = 0.f;
        for (int r = 0; r < 512; ++r) acc = fmaf(W1[n * 512 + r], Wsr[r * 4 + o], acc);
        Wc[tid] = acc;
    } else if (tid < 132) {                // bc[o]
        int o = tid - 128;
        float acc = br[o];
        for (int r = 0; r < 512; ++r)  acc = fmaf(b1[r], Wsr[r * 4 + o], acc);
        for (int k = 0; k < 4096; ++k) acc = fmaf(bs[k], Wr[k * 4 + o], acc);
        bc[o] = acc;
    }
}

// ---------------- Stage 1: window121 -> fp32 WMMA (484->32) -> tanh -> (32->4) ----------------
__global__ __launch_bounds__(256) void k_stage1(const float* __restrict__ x,
                                                const float* __restrict__ W0,
                                                const float* __restrict__ b0,
                                                const float* __restrict__ Wc,
                                                const float* __restrict__ bc,
                                                float* __restrict__ hmid) {
    __shared__ float xt[(TB + HALO) * 4];     // 992 floats: x halo tile [time][c]
    __shared__ float w0t[H0 * K0];            // 15488 floats: W0 transposed [n][k]
    __shared__ float b0s[H0];
    __shared__ float wcs[H0 * 4];
    __shared__ float bcs[4];
    __shared__ float h0s[TB * 36];            // stride 36 to dodge bank conflicts

    const int tid = threadIdx.x;
    const int bb  = blockIdx.x >> 5;          // batch (32 tiles per batch)
    const int t0  = (blockIdx.x & 31) * TB;   // first token of this tile

    // cooperative LDS fills
    for (int i = tid; i < (TB + HALO) * 4; i += 256) {
        int j = i >> 2, c = i & 3;
        int gt = (t0 + j + L_TOK - HALO) & (L_TOK - 1);     // circular
        xt[i] = x[((bb << 12) + gt) * 4 + c];
    }
    for (int i = tid; i < H0 * K0; i += 256) {              // W0 is [k][n] row-major
        int k = i >> 5, n = i & 31;
        w0t[n * K0 + k] = W0[i];
    }
    if (tid < H0)  b0s[tid] = b0[tid];
    if (tid < 128) wcs[tid] = Wc[tid];
    if (tid < 4)   bcs[tid] = bc[tid];
    __syncthreads();

    const int wave = tid >> 5;
    const int lane = tid & 31;
    const int hl   = lane >> 4;        // half-wave: selects K 0/1 vs 2/3
    const int m0   = lane & 15;        // row for A, col for B
    const int c0   = hl << 1;

    v8f acc0 = {}; v8f acc1 = {};
    const float* ap  = xt  + (((wave << 4) + m0)) * 4 + c0;   // A: token (wave*16+m0), k-chunk base
    const float* bp0 = w0t + m0 * K0 + c0;                    // B tile0: n = m0
    const float* bp1 = w0t + (m0 + 16) * K0 + c0;             // B tile1: n = m0+16

    for (int kk = 0; kk < 121; ++kk) {
        v2f a  = *(const v2f*)(ap  + 4 * kk);   // window j=kk, channels c0,c0+1
        v2f bA = *(const v2f*)(bp0 + 4 * kk);   // W0[4kk+c0..+1][n]
        v2f bB = *(const v2f*)(bp1 + 4 * kk);
        acc0 = __builtin_amdgcn_wmma_f32_16x16x4_f32(false, a, false, bA, (short)0, acc0, false, false);
        acc1 = __builtin_amdgcn_wmma_f32_16x16x4_f32(false, a, false, bB, (short)0, acc1, false, false);
    }

    // bias + tanh, scatter D-layout (VGPR g = rows g / g+8) into LDS
    #pragma unroll
    for (int g = 0; g < 8; ++g) {
        int M = (wave << 4) + g + (hl << 3);
        h0s[M * 36 + m0]      = tanhf(acc0[g] + b0s[m0]);
        h0s[M * 36 + m0 + 16] = tanhf(acc1[g] + b0s[m0 + 16]);
    }
    __syncthreads();

    // collapsed middle: hmid[tok][o] = h0[tok][:] . Wc[:,o] + bc[o]
    #pragma unroll
    for (int r = 0; r < 2; ++r) {
        int idx = tid + (r << 8);
        int tok = idx >> 2, o = idx & 3;
        const float* hrow = h0s + tok * 36;
        float s = bcs[o];
        #pragma unroll 8
        for (int n = 0; n < 32; ++n) s = fmaf(hrow[n], wcs[n * 4 + o], s);
        hmid[(((bb << 12) + t0 + tok) << 2) + o] = s;
    }
}

// ---------------- Stage 2: window5 -> relu(20->32) -> (32->4) ----------------
__global__ __launch_bounds__(256) void k_stage2(const float* __restrict__ hmid,
                                                const float* __restrict__ W2,
                                                const float* __restrict__ b2,
                                                const float* __restrict__ W3,
                                                const float* __restrict__ b3,
                                                float* __restrict__ out) {
    __shared__ float w2s[20 * 32];
    __shared__ float b2s[32];
    __shared__ float w3s[32 * 4];
    __shared__ float b3s[4];
    int tid = threadIdx.x;
    for (int i = tid; i < 640; i += 256) w2s[i] = W2[i];
    if (tid < 32)  b2s[tid] = b2[tid];
    if (tid < 128) w3s[tid] = W3[tid];
    if (tid < 4)   b3s[tid] = b3[tid];
    __syncthreads();

    int g  = blockIdx.x * 256 + tid;           // global token
    int bb = g >> 12, t = g & (L_TOK - 1);

    float win[20];
    #pragma unroll
    for (int j = 0; j < 5; ++j) {
        int tt = (t + L_TOK - 4 + j) & (L_TOK - 1);
        const float* p = hmid + (((bb << 12) + tt) << 2);
        win[j * 4 + 0] = p[0]; win[j * 4 + 1] = p[1];
        win[j * 4 + 2] = p[2]; win[j * 4 + 3] = p[3];
    }
    float o0 = b3s[0], o1 = b3s[1], o2 = b3s[2], o3 = b3s[3];
    for (int n = 0; n < 32; ++n) {
        float h = b2s[n];
        #pragma unroll
        for (int k = 0; k < 20; ++k) h = fmaf(win[k], w2s[k * 32 + n], h);
        h = fmaxf(h, 0.f);
        o0 = fmaf(h, w3s[n * 4 + 0], o0);
        o1 = fmaf(h, w3s[n * 4 + 1], o1);
        o2 = fmaf(h, w3s[n * 4 + 2], o2);
        o3 = fmaf(h, w3s[n * 4 + 3], o3);
    }
    float* q = out + ((size_t)g << 2);
    q[0] = o0; q[1] = o1; q[2] = o2; q[3] = o3;
}

extern "C" void kernel_launch(void* const* d_in, const int* in_sizes, int n_in,
                              void* d_out, int out_size, void* d_ws, size_t ws_size,
                              hipStream_t stream) {
    (void)in_sizes; (void)n_in; (void)out_size; (void)ws_size;
    const float* x  = (const float*)d_in[0];
    const float* W0 = (const float*)d_in[1];
    const float* b0 = (const float*)d_in[2];
    const float* W1 = (const float*)d_in[3];
    const float* b1 = (const float*)d_in[4];
    const float* Ws = (const float*)d_in[5];
    const float* bs = (const float*)d_in[6];
    const float* Wr = (const float*)d_in[7];
    const float* br = (const float*)d_in[8];
    const float* W2 = (const float*)d_in[9];
    const float* b2 = (const float*)d_in[10];
    const float* W3 = (const float*)d_in[11];
    const float* b3 = (const float*)d_in[12];
    float* out = (float*)d_out;

    // workspace layout (floats): Wsr[2048] | Wc[128] | bc[4] | pad | hmid[B*L*4]
    float* ws   = (float*)d_ws;
    float* Wsr  = ws;
    float* Wc   = ws + 2048;
    float* bc   = ws + 2176;
    float* hmid = ws + 2304;

    k_wsr   <<<8,   256, 0, stream>>>(Ws, Wr, Wsr);
    k_fold  <<<1,   256, 0, stream>>>(W1, b1, bs, Wr, br, Wsr, Wc, bc);
    k_stage1<<<256, 256, 0, stream>>>(x, W0, b0, Wc, bc, hmid);
    k_stage2<<<128, 256, 0, stream>>>(hmid, W2, b2, W3, b3, out);
}